// RelationalAttentionRGCN_35948876268066
// MI455X (gfx1250) — compile-verified
//
#include <hip/hip_runtime.h>

#define DEV_INLINE __device__ __forceinline__

typedef float v2f __attribute__((ext_vector_type(2)));
typedef float v8f __attribute__((ext_vector_type(8)));
typedef unsigned v4u __attribute__((ext_vector_type(4)));
typedef int v4i __attribute__((ext_vector_type(4)));
typedef int v8i __attribute__((ext_vector_type(8)));

constexpr int BSZ  = 4096;   // batch
constexpr int KNEI = 64;     // neighbors
constexpr int DIM  = 128;    // embed dim
constexpr int NREL = 16;     // relations

// D = A(16x4) * B(4x16) + C, fp32 in / fp32 out (exact match to reference math)
DEV_INLINE v8f wmma_f32_k4(v2f a, v2f b, v8f c) {
  return __builtin_amdgcn_wmma_f32_16x16x4_f32(false, a, false, b, (short)0, c,
                                               false, false);
}

// Flat LDS addresses keep the workgroup-relative offset in the low 32 bits
// (ISA 10.2), so truncation yields the DS offset.
DEV_INLINE unsigned lds_off(const void* p) { return (unsigned)(size_t)p; }

// Async global->LDS 16B copy, per-lane addresses, tracked by ASYNCcnt.
DEV_INLINE void async_ld16(unsigned lds, const void* g) {
  asm volatile("global_load_async_to_lds_b128 %0, %1, off"
               :: "v"(lds), "v"(g) : "memory");
}
DEV_INLINE void wait_async0()  { asm volatile("s_wait_asynccnt 0x0" ::: "memory"); }
DEV_INLINE void wait_tensor0() { __builtin_amdgcn_s_wait_tensorcnt(0); }

// TDM: DMA a 2D fp32 tile (tile_w x tile_h elems, global row stride
// g_row_stride elems) into LDS, optionally inserting LDS padding every
// 2^(pad_int_code+1) dwords of pad_amt_code+1 dwords (-> padded row stride).
DEV_INLINE void tdm_load_2d(unsigned lds_addr, const void* gaddr,
                            unsigned tile_w, unsigned tile_h,
                            unsigned g_row_stride, bool pad_en,
                            unsigned pad_int_code, unsigned pad_amt_code) {
  const unsigned long long ga = (unsigned long long)(size_t)gaddr;
  v4u g0;
  g0.x = 1u;                                              // count=1, user mode
  g0.y = lds_addr;                                        // LDS byte address
  g0.z = (unsigned)ga;                                    // global_addr[31:0]
  g0.w = (unsigned)((ga >> 32) & 0x01ffffffu) | (2u << 30);  // ga[56:32]|type=2
  unsigned w0 = (2u << 16);                               // data_size = 4B
  if (pad_en) w0 |= (1u << 20) | (pad_int_code << 22) | (pad_amt_code << 25);
  v8i g1;
  g1[0] = (int)w0;                                        // wg_mask=0, flags
  g1[1] = (int)((tile_w & 0xffffu) << 16);                // tensor_dim0 lo16
  g1[2] = (int)(((tile_w >> 16) & 0xffffu) |
                ((tile_h & 0xffffu) << 16));              // td0 hi | td1 lo
  g1[3] = (int)(((tile_h >> 16) & 0xffffu) |
                ((tile_w & 0xffffu) << 16));              // td1 hi | tile_dim0
  g1[4] = (int)(tile_h & 0xffffu);                        // tile_dim1; tile_dim2=0
  g1[5] = (int)g_row_stride;                              // tensor_dim0_stride
  g1[6] = 0;
  g1[7] = 0;
  v4i g2 = {0, 0, 0, 0};                                  // dims 2/3 unused
  v4i g3 = {0, 0, 0, 0};
#if defined(__clang_major__) && (__clang_major__ >= 23)
  v8i gz = {0, 0, 0, 0, 0, 0, 0, 0};
  __builtin_amdgcn_tensor_load_to_lds(g0, g1, g2, g3, gz, 0);
#else
  __builtin_amdgcn_tensor_load_to_lds(g0, g1, g2, g3, 0);
#endif
}

// ---------------------------------------------------------------------------
// Kernel 1: gather x0 = E[drug_idx]; agg[b,r,:] = sum_{k: rel=r} w*E[adj[b,k]]
// One wave per batch row; 16x128 fp32 accumulator slab in LDS (8 KB/wave).
// E (51 MB) is L2-resident; prefetch next row to hide the dependent gather.
// ---------------------------------------------------------------------------
__global__ __launch_bounds__(256) void k_gather_agg(
    const int* __restrict__ drug, const int* __restrict__ adjE,
    const int* __restrict__ adjR, const float* __restrict__ ew,
    const float* __restrict__ E, float* __restrict__ agg,
    float* __restrict__ x0) {
  __shared__ float sm[8][NREL * DIM];   // 64 KB total
  const int wave = threadIdx.x >> 5;
  const int lane = threadIdx.x & 31;
  const int b    = blockIdx.x * 8 + wave;
  float* acc = sm[wave];
  for (int i = lane; i < NREL * DIM; i += 32) acc[i] = 0.0f;
  {
    const int e = drug[b];
    const float4 v = reinterpret_cast<const float4*>(E + (size_t)e * DIM)[lane];
    reinterpret_cast<float4*>(x0 + (size_t)b * DIM)[lane] = v;
  }
  int e_next = adjE[b * KNEI];
  for (int k = 0; k < KNEI; ++k) {
    const int e = e_next;
    if (k + 1 < KNEI) {
      e_next = adjE[b * KNEI + k + 1];
      __builtin_prefetch(E + (size_t)e_next * DIM + lane * 4, 0, 1);
    }
    const int   r = adjR[b * KNEI + k];
    const float w = ew[b * KNEI + k];
    const float4 v = reinterpret_cast<const float4*>(E + (size_t)e * DIM)[lane];
    float* dst = acc + r * DIM + lane * 4;   // lane owns 4 floats; wave-private
    dst[0] += w * v.x; dst[1] += w * v.y; dst[2] += w * v.z; dst[3] += w * v.w;
  }
  float4* out4 = reinterpret_cast<float4*>(agg + (size_t)b * NREL * DIM);
  const float4* s4 = reinterpret_cast<const float4*>(acc);
  for (int i = lane; i < NREL * DIM / 4; i += 32) out4[i] = s4[i];
}

// ---------------------------------------------------------------------------
// WMMA inner chunk (row-major B, used with the TDM-staged tiles):
// A-frag (f32 16x4): lane<16 -> row=lane,{K0,K1}; lane>=16 -> row,{K2,K3}
//   -> contiguous 8B in LDS => single ds_load_b64 into an aligned pair.
// B-frag (f32 4x16): v0 = rows {0,2}, v1 = rows {1,3} striped across lanes.
// ---------------------------------------------------------------------------
DEV_INLINE void mma_chunk(const float* __restrict__ sArow, int a_stride,
                          const float* __restrict__ sBbase, int b_stride,
                          int lane, v8f acc[8]) {
  const int m    = lane & 15;
  const int koff = (lane < 16) ? 0 : 2;
  #pragma unroll
  for (int kk = 0; kk < 64; kk += 4) {
    const v2f a =
        *reinterpret_cast<const v2f*>(sArow + m * a_stride + kk + koff);
    #pragma unroll
    for (int nt = 0; nt < 8; ++nt) {
      v2f bb;
      bb.x = sBbase[(kk + koff) * b_stride + nt * 16 + m];
      bb.y = sBbase[(kk + koff + 1) * b_stride + nt * 16 + m];
      acc[nt] = wmma_f32_k4(a, bb, acc[nt]);
    }
  }
}

// ---------------------------------------------------------------------------
// Kernel 2: one RGCN layer, fully fused:
//   x_out = relu( x_in + Agg[B,2048] @ Wcat[2048,128] + x_in @ res_W^T + res_b )
// Block = 4 waves; wave owns a 16-row M-tile. Double-buffered K-chunk
// pipeline: B tile via TDM (wave 0, TENSORcnt, LDS padding 128->144 dwords),
// A tile via global_load_async_to_lds_b128 (ASYNCcnt).
// ---------------------------------------------------------------------------
__global__ __launch_bounds__(128) void k_rgcn_layer(
    const float* __restrict__ agg, const float* __restrict__ x_in,
    const float* __restrict__ Wcat, const float* __restrict__ resW,
    const float* __restrict__ resb, float* __restrict__ x_out) {
  constexpr int KC  = 64;
  constexpr int BST = 144;  // B tile row stride (floats) = 128 + 16 TDM pad
  constexpr int AST = 68;   // A tile row stride (floats), 272B = 17*16B
  constexpr int NCHUNK = (NREL * DIM) / KC;  // 32
  __shared__ __align__(16) float sB[2][KC * BST];    // 2 x 36 KB
  __shared__ __align__(16) float sA[2][4][16 * AST]; // 2 x 17 KB
  const int wave = threadIdx.x >> 5;
  const int lane = threadIdx.x & 31;
  const int m0   = blockIdx.x * 64 + wave * 16;

  v8f acc[8] = {};

  auto issueB = [&](int c, int buf) {   // whole 64x128 chunk via one TDM op
    if (wave == 0)
      tdm_load_2d(lds_off(&sB[buf][0]), Wcat + (size_t)(c * KC) * DIM,
                  /*tile_w=*/DIM, /*tile_h=*/KC, /*stride=*/DIM,
                  /*pad_en=*/true, /*2^(6+1)=128 dw*/ 6u, /*16 dw*/ 15u);
  };
  auto issueA = [&](int c, int buf) {   // per-lane async 16B copies
    const float* src = agg + (size_t)m0 * (NREL * DIM) + c * KC;
    for (int i = lane; i < 16 * 16; i += 32) {
      const int row = i >> 4, c4 = i & 15;
      async_ld16(lds_off(&sA[buf][wave][row * AST + c4 * 4]),
                 src + (size_t)row * (NREL * DIM) + c4 * 4);
    }
  };

  issueB(0, 0);
  issueA(0, 0);
  wait_async0();
  if (wave == 0) wait_tensor0();
  __syncthreads();

  for (int c = 0; c < NCHUNK; ++c) {
    const int buf = c & 1;
    if (c + 1 < NCHUNK) { issueB(c + 1, buf ^ 1); issueA(c + 1, buf ^ 1); }
    mma_chunk(sA[buf][wave], AST, sB[buf], BST, lane, acc);
    wait_async0();
    if (wave == 0) wait_tensor0();
    __syncthreads();
  }

  // ---- residual GEMM: x_in[16 x 128] x res_W^T (transpose during staging)
  for (int k0 = 0; k0 < DIM; k0 += KC) {
    for (int i = threadIdx.x; i < DIM * 16; i += 128) {  // 128 e-rows x 16 f4
      const int e = i >> 4, c4 = i & 15;
      const float4 v =
          reinterpret_cast<const float4*>(resW + (size_t)e * DIM + k0)[c4];
      sB[0][(c4 * 4 + 0) * BST + e] = v.x;
      sB[0][(c4 * 4 + 1) * BST + e] = v.y;
      sB[0][(c4 * 4 + 2) * BST + e] = v.z;
      sB[0][(c4 * 4 + 3) * BST + e] = v.w;
    }
    for (int i = lane; i < 16 * 16; i += 32) {
      const int row = i >> 4, c4 = i & 15;
      async_ld16(lds_off(&sA[0][wave][row * AST + c4 * 4]),
                 x_in + (size_t)(m0 + row) * DIM + k0 + c4 * 4);
    }
    wait_async0();
    __syncthreads();
    mma_chunk(sA[0][wave], AST, sB[0], BST, lane, acc);
    __syncthreads();
  }

  // ---- epilogue: relu(x_in + msgs + resid + res_b)
  const int col0  = lane & 15;
  const int rbase = (lane < 16) ? 0 : 8;   // C layout: vgpr j -> rows j, j+8
  #pragma unroll
  for (int nt = 0; nt < 8; ++nt) {
    #pragma unroll
    for (int j = 0; j < 8; ++j) {
      const int row = m0 + rbase + j;
      const int col = nt * 16 + col0;
      const float v = acc[nt][j] + x_in[(size_t)row * DIM + col] + resb[col];
      x_out[(size_t)row * DIM + col] = fmaxf(v, 0.0f);
    }
  }
}

// ---------------------------------------------------------------------------
// Kernel 3/5: generic C[M,N] = A[M,128] @ W[N,128]^T + bias   (K fixed = 128)
// A rows staged once per wave via async-to-LDS, then all 32 A-fragments are
// preloaded into registers. W blocks are staged PAIR-INTERLEAVED:
//   element (k,n) -> sB[(k>>1)*34 + 2n + (k&1)]
// so the (k,k+1) couple each lane needs is one aligned 8B ds_load_b64 ->
// even VGPR pair feeding WMMA directly (no v_mov repacking).
// ---------------------------------------------------------------------------
template <int NCOLS>
__global__ __launch_bounds__(128) void k_gemm_xWT(
    const float* __restrict__ A, const float* __restrict__ W,
    const float* __restrict__ bias, float* __restrict__ C, int M) {
  constexpr int AST = 132;  // 528B rows = 33*16B (aligned for b128 async)
  constexpr int PST = 34;   // paired-B row stride: 16 pairs * 2 + 2 pad
  __shared__ __align__(16) float sA[4][16 * AST];      // 33.8 KB
  __shared__ __align__(16) float sB[(DIM / 2) * PST];  // 8.7 KB
  const int wave = threadIdx.x >> 5;
  const int lane = threadIdx.x & 31;
  const int m0   = blockIdx.x * 64 + wave * 16;
  (void)M;  // grid sized so every tile is full

  for (int i = lane; i < 16 * 32; i += 32) {   // async-stage full-K A rows
    const int row = i >> 5, c4 = i & 31;
    async_ld16(lds_off(&sA[wave][row * AST + c4 * 4]),
               A + (size_t)(m0 + row) * DIM + c4 * 4);
  }

  const int m     = lane & 15;
  const int koffp = (lane < 16) ? 0 : 1;   // pair-row parity (K0K1 vs K2K3)
  const int rbase = (lane < 16) ? 0 : 8;

  wait_async0();
  v2f afr[DIM / 4];                        // all A fragments, kept in VGPRs
  #pragma unroll
  for (int s = 0; s < DIM / 4; ++s)
    afr[s] =
        *reinterpret_cast<const v2f*>(&sA[wave][m * AST + 4 * s + 2 * koffp]);

  for (int nt = 0; nt < NCOLS / 16; ++nt) {
    __syncthreads();   // protect sB reuse across iterations
    for (int i = threadIdx.x; i < 16 * 32; i += 128) {  // W block, interleaved
      const int n = i >> 5, c4 = i & 31;
      const float4 v = reinterpret_cast<const float4*>(
          W + (size_t)(nt * 16 + n) * DIM)[c4];
      const int kp = c4 * 2;               // k = 4*c4 -> pair rows kp, kp+1
      v2f lo; lo.x = v.x; lo.y = v.y;
      v2f hi; hi.x = v.z; hi.y = v.w;
      *reinterpret_cast<v2f*>(&sB[(kp + 0) * PST + 2 * n]) = lo;
      *reinterpret_cast<v2f*>(&sB[(kp + 1) * PST + 2 * n]) = hi;
    }
    __syncthreads();
    v8f acc = {};
    #pragma unroll
    for (int s = 0; s < DIM / 4; ++s) {
      const v2f bb = *reinterpret_cast<const v2f*>(
          &sB[(2 * s + koffp) * PST + 2 * m]);
      acc = wmma_f32_k4(afr[s], bb, acc);
    }
    #pragma unroll
    for (int j = 0; j < 8; ++j) {
      const int row = m0 + rbase + j;
      const int col = nt * 16 + m;
      C[(size_t)row * NCOLS + col] = acc[j] + bias[col];
    }
  }
}

// ---------------------------------------------------------------------------
// Kernel 4: tiny L=2 attention per batch row; head = wave (dh = 32 = wave32).
// Folds the mean over L through the (linear) out_proj: o_avg = (o0+o1)/2.
// ---------------------------------------------------------------------------
__global__ __launch_bounds__(128) void k_attn(const float* __restrict__ qkv,
                                              float* __restrict__ oavg) {
  const int b = blockIdx.x;
  const int d = threadIdx.x;           // 0..127 ; head = d>>5 = wave id
  const float* r0 = qkv + (size_t)b * (3 * DIM);
  const float* r1 = qkv + (size_t)(BSZ + b) * (3 * DIM);
  const float q0 = r0[d], k0 = r0[DIM + d], v0 = r0[2 * DIM + d];
  const float q1 = r1[d], k1 = r1[DIM + d], v1 = r1[2 * DIM + d];
  float p00 = q0 * k0, p01 = q0 * k1, p10 = q1 * k0, p11 = q1 * k1;
  #pragma unroll
  for (int off = 16; off > 0; off >>= 1) {   // wave32 butterfly reduce
    p00 += __shfl_xor(p00, off, 32);
    p01 += __shfl_xor(p01, off, 32);
    p10 += __shfl_xor(p10, off, 32);
    p11 += __shfl_xor(p11, off, 32);
  }
  const float scale = 0.17677669529663687f;  // 1/sqrt(32)
  const float s00 = p00 * scale, s01 = p01 * scale;
  const float s10 = p10 * scale, s11 = p11 * scale;
  const float m0s = fmaxf(s00, s01);
  const float e00 = __expf(s00 - m0s), e01 = __expf(s01 - m0s);
  const float a00 = e00 / (e00 + e01), a01 = e01 / (e00 + e01);
  const float m1s = fmaxf(s10, s11);
  const float e10 = __expf(s10 - m1s), e11 = __expf(s11 - m1s);
  const float a10 = e10 / (e10 + e11), a11 = e11 / (e10 + e11);
  const float o0 = a00 * v0 + a01 * v1;
  const float o1 = a10 * v0 + a11 * v1;
  oavg[(size_t)b * DIM + d] = 0.5f * (o0 + o1);
}

// ---------------------------------------------------------------------------
extern "C" void kernel_launch(void* const* d_in, const int* in_sizes, int n_in,
                              void* d_out, int out_size, void* d_ws,
                              size_t ws_size, hipStream_t stream) {
  (void)in_sizes; (void)n_in; (void)out_size; (void)ws_size;
  const int*   drug = (const int*)d_in[0];
  const int*   adjE = (const int*)d_in[1];
  const int*   adjR = (const int*)d_in[2];
  const float* ew   = (const float*)d_in[3];
  const float* E    = (const float*)d_in[4];
  const float* Wrel = (const float*)d_in[5];   // [2,16,128,128] == [2][2048,128]
  const float* resW = (const float*)d_in[6];   // [2,128,128]
  const float* resb = (const float*)d_in[7];   // [2,128]
  const float* inW  = (const float*)d_in[8];   // [384,128]
  const float* inB  = (const float*)d_in[9];   // [384]
  const float* outW = (const float*)d_in[10];  // [128,128]
  const float* outB = (const float*)d_in[11];  // [128]
  float* out = (float*)d_out;                  // [4096,128]

  float* ws   = (float*)d_ws;
  float* agg  = ws;                               // B*16*128  = 33.6 MB
  float* x0   = agg  + (size_t)BSZ * NREL * DIM;  // B*128     =  2.1 MB
  float* xL   = x0   + (size_t)BSZ * DIM;         // [2][B][128] (layer outs)
  float* qkv  = xL   + (size_t)2 * BSZ * DIM;     // [2B][384] = 12.6 MB
  float* oavg = qkv  + (size_t)2 * BSZ * 3 * DIM; // [B][128]

  k_gather_agg<<<BSZ / 8, 256, 0, stream>>>(drug, adjE, adjR, ew, E, agg, x0);
  k_rgcn_layer<<<BSZ / 64, 128, 0, stream>>>(agg, x0, Wrel, resW, resb, xL);
  k_rgcn_layer<<<BSZ / 64, 128, 0, stream>>>(
      agg, xL, Wrel + (size_t)NREL * DIM * DIM, resW + (size_t)DIM * DIM,
      resb + DIM, xL + (size_t)BSZ * DIM);
  k_gemm_xWT<3 * DIM><<<(2 * BSZ) / 64, 128, 0, stream>>>(xL, inW, inB, qkv,
                                                          2 * BSZ);
  k_attn<<<BSZ, 128, 0, stream>>>(qkv, oavg);
  k_gemm_xWT<DIM><<<BSZ / 64, 128, 0, stream>>>(oavg, outW, outB, out, BSZ);
}